// Generator_77034533421677
// MI455X (gfx1250) — compile-verified
//
#include <hip/hip_runtime.h>
#include <math.h>

typedef __attribute__((ext_vector_type(2))) float v2f;
typedef __attribute__((ext_vector_type(8))) float v8f;

#define IN_EPS 1e-5f

// Network: (Cin,Cout,k,s,p), spatial per sample:
// L1: 100->128 k4 s2 p1 : 1x1  -> 2x2
// L2: 128->64  k4 s2 p1 : 2x2  -> 4x4
// L3:  64->32  k3 s2 p1 : 4x4  -> 7x7
// L4:  32->16  k4 s2 p1 : 7x7  -> 14x14
// L5:  16->1   k4 s2 p1 : 14x14-> 28x28
//
// Adder transposed conv rewritten:
//   out[co,ho,wo] = -( S[co] + sum_{real taps} (|x - wf| - |wf|) )
// where S[co] = sum_{ci,i,j} |w[ci,co,i,j]| (zero-dilation/padding terms),
// wf[co,ci,i,j] = w[ci,co,K-1-i,K-1-j].

// ---------------- pass 0: per-layer per-cout sum of |w| into d_ws ----------
__global__ void precompute_wabs(const float* __restrict__ w1, const float* __restrict__ w2,
                                const float* __restrict__ w3, const float* __restrict__ w4,
                                const float* __restrict__ w5, float* __restrict__ S) {
  int t = threadIdx.x;
  const float* w; int CIN, COUT, KK, co, off;
  if (t < 128)       { w = w1; CIN = 100; COUT = 128; KK = 16; co = t;       off = 0;   }
  else if (t < 192)  { w = w2; CIN = 128; COUT = 64;  KK = 16; co = t - 128; off = 128; }
  else if (t < 224)  { w = w3; CIN = 64;  COUT = 32;  KK = 9;  co = t - 192; off = 192; }
  else if (t < 240)  { w = w4; CIN = 32;  COUT = 16;  KK = 16; co = t - 224; off = 224; }
  else if (t == 240) { w = w5; CIN = 16;  COUT = 1;   KK = 16; co = 0;       off = 240; }
  else return;
  float acc = 0.0f;
  for (int ci = 0; ci < CIN; ++ci)
    for (int e = 0; e < KK; ++e)
      acc += fabsf(w[(ci * COUT + co) * KK + e]);
  S[off + co] = acc;
}

// ---------------- fused layer: adder t-conv + instance norm + activation ---
template <int CIN, int COUT, int K, int STR, int P,
          int HIN, int WIN, int HOUT, int WOUT, bool TANH>
__device__ __forceinline__ void adder_layer(const float* __restrict__ w,
                                            const float* __restrict__ Sc,
                                            const float* hin, float* hout,
                                            float* sums, float* sumsq) {
  constexpr int PAD  = K - 1 - P;
  constexpr int NPIX = HOUT * WOUT;
  constexpr int NOUT = COUT * NPIX;
  const int tid = threadIdx.x;

  // Raw adder transposed conv (VALU: L1 distance is not bilinear -> no WMMA here)
  for (int idx = tid; idx < NOUT; idx += 256) {
    const int co = idx / NPIX;
    const int r  = idx - co * NPIX;
    const int ho = r / WOUT;
    const int wo = r - ho * WOUT;
    float acc = Sc[co];
    #pragma unroll
    for (int i = 0; i < K; ++i) {
      const int hz = ho + i - PAD;
      if (hz < 0) continue;
      if (hz % STR) continue;
      const int hx = hz / STR;
      if (hx >= HIN) continue;
      #pragma unroll
      for (int j = 0; j < K; ++j) {
        const int wz = wo + j - PAD;
        if (wz < 0) continue;
        if (wz % STR) continue;
        const int wx = wz / STR;
        if (wx >= WIN) continue;
        const int widx  = co * (K * K) + (K - 1 - i) * K + (K - 1 - j);
        const int xbase = hx * WIN + wx;
        for (int ci = 0; ci < CIN; ++ci) {
          const float wv = w[ci * (COUT * K * K) + widx];
          const float xv = hin[ci * (HIN * WIN) + xbase];
          acc += fabsf(xv - wv) - fabsf(wv);
        }
      }
    }
    hout[idx] = -acc;
  }
  __syncthreads();

  // Instance-norm reductions via WMMA f32 16x16x4 with ones-B (exact f32 row sums).
  // Wave 0 only; branch is wave-uniform so EXEC is all ones at the WMMA.
  if (tid < 32) {
    const int lane = tid;
    const int m  = lane & 15;                // A-matrix row (channel within tile)
    const int kb = (lane < 16) ? 0 : 2;      // K slice held by this lane half
    v2f ones; ones.x = 1.0f; ones.y = 1.0f;
    for (int ct = 0; ct < COUT; ct += 16) {
      v8f accS = {};
      v8f accQ = {};
      const int  ch   = ct + m;
      const bool chok = (ch < COUT);
      for (int p0 = 0; p0 < NPIX; p0 += 4) {
        float a0 = 0.0f, a1 = 0.0f;
        if (chok) {
          const int pp = p0 + kb;
          if (pp < NPIX)     a0 = hout[ch * NPIX + pp];
          if (pp + 1 < NPIX) a1 = hout[ch * NPIX + pp + 1];
        }
        v2f a;  a.x  = a0;      a.y  = a1;
        v2f aq; aq.x = a0 * a0; aq.y = a1 * a1;
        // D[m][n] += sum_k A[m,k] * 1  -> row sums accumulate in f32
        accS = __builtin_amdgcn_wmma_f32_16x16x4_f32(false, a,  false, ones,
                                                     (short)0, accS, false, false);
        accQ = __builtin_amdgcn_wmma_f32_16x16x4_f32(false, aq, false, ones,
                                                     (short)0, accQ, false, false);
      }
      // C/D layout: VGPR r, lanes 0-15 hold M=r, lanes 16-31 hold M=r+8.
      if (m == 0) {
        const int cb = ct + ((lane < 16) ? 0 : 8);
        #pragma unroll
        for (int r = 0; r < 8; ++r) {
          if (cb + r < COUT) {
            sums[cb + r]  = accS[r];
            sumsq[cb + r] = accQ[r];
          }
        }
      }
    }
  }
  __syncthreads();

  // Normalize + activation (in place)
  const float invN = 1.0f / (float)NPIX;
  for (int idx = tid; idx < NOUT; idx += 256) {
    const int co = idx / NPIX;
    const float mean = sums[co] * invN;
    const float var  = sumsq[co] * invN - mean * mean;
    const float v    = (hout[idx] - mean) * rsqrtf(var + IN_EPS);
    hout[idx] = TANH ? tanhf(v) : fmaxf(v, 0.0f);
  }
  __syncthreads();
}

// ---------------- pass 1: whole network fused, one block per sample --------
__global__ void __launch_bounds__(256)
addergen_fused(const float* __restrict__ x,
               const float* __restrict__ w1, const float* __restrict__ w2,
               const float* __restrict__ w3, const float* __restrict__ w4,
               const float* __restrict__ w5,
               const float* __restrict__ Sall,
               float* __restrict__ out) {
  __shared__ float bufA[3136];   // max intermediate: 16*14*14
  __shared__ float bufB[3136];
  __shared__ float sums[128];
  __shared__ float sumsq[128];

  const int b   = blockIdx.x;
  const int tid = threadIdx.x;

  // load latent x[b] (100 floats, H=W=1)
  for (int i = tid; i < 100; i += 256) bufA[i] = x[b * 100 + i];
  __syncthreads();

  adder_layer<100, 128, 4, 2, 1,  1,  1,  2,  2, false>(w1, Sall + 0,   bufA, bufB, sums, sumsq);
  adder_layer<128,  64, 4, 2, 1,  2,  2,  4,  4, false>(w2, Sall + 128, bufB, bufA, sums, sumsq);
  adder_layer< 64,  32, 3, 2, 1,  4,  4,  7,  7, false>(w3, Sall + 192, bufA, bufB, sums, sumsq);
  adder_layer< 32,  16, 4, 2, 1,  7,  7, 14, 14, false>(w4, Sall + 224, bufB, bufA, sums, sumsq);
  adder_layer< 16,   1, 4, 2, 1, 14, 14, 28, 28, true >(w5, Sall + 240, bufA, bufB, sums, sumsq);

  for (int i = tid; i < 784; i += 256) out[b * 784 + i] = bufB[i];
}

extern "C" void kernel_launch(void* const* d_in, const int* in_sizes, int n_in,
                              void* d_out, int out_size, void* d_ws, size_t ws_size,
                              hipStream_t stream) {
  const float* x  = (const float*)d_in[0];
  const float* w1 = (const float*)d_in[1];
  const float* w2 = (const float*)d_in[2];
  const float* w3 = (const float*)d_in[3];
  const float* w4 = (const float*)d_in[4];
  const float* w5 = (const float*)d_in[5];
  float* S = (float*)d_ws;   // 241 floats: S1[128] S2[64] S3[32] S4[16] S5[1]

  precompute_wabs<<<1, 256, 0, stream>>>(w1, w2, w3, w4, w5, S);
  addergen_fused<<<512, 256, 0, stream>>>(x, w1, w2, w3, w4, w5, S, (float*)d_out);
}